// GATLayer_55765855371638
// MI455X (gfx1250) — compile-verified
//
#include <hip/hip_runtime.h>
#include <hip/hip_bf16.h>
#include <math.h>

#define IN_DIM   128
#define OUT_DIM  256
#define HEADS    8
#define CH       32
#define NEG_SLOPE 0.2f
#define LN_EPS    1e-5f

#define LDS_PITCH (IN_DIM + 2)   // TDM pads 2 DWORDs per 128-DWORD row (bank decollision)

typedef __attribute__((ext_vector_type(2))) float v2f;
typedef __attribute__((ext_vector_type(8))) float v8f;
typedef __attribute__((ext_vector_type(4))) unsigned int u32x4;
typedef __attribute__((ext_vector_type(8))) int i32x8;
typedef __attribute__((ext_vector_type(4))) int i32x4;

// -------------------------------------------------------------------------
// Fused fp32 WMMA GEMM: out[M x 256] = A[M x 128] * B
// B element (k, o) = Bsrc[k*strideK + o*strideO]
//   - for W      [128,256]: strideK=256, strideO=1
//   - for res_W^T[128,256] : strideK=1,  strideO=128  (res_W is [256,128])
// Block = 256 threads = 8 waves; wave w computes col tiles {2w, 2w+1} of a
// 16-row stripe. The 16x128 A tile is staged into LDS once per block by the
// Tensor Data Mover (with row padding), then each wave runs 64 fp32 WMMAs.
// -------------------------------------------------------------------------
__global__ __launch_bounds__(256) void gat_gemm_f32_wmma(
    const float* __restrict__ A, const float* __restrict__ Bsrc,
    float* __restrict__ out, int M, int strideK, int strideO)
{
    __shared__ float lds_A[16 * LDS_PITCH];

    const int lane = threadIdx.x & 31;
    const int wave = threadIdx.x >> 5;          // 0..7
    const int m0   = blockIdx.x * 16;
    if (m0 >= M) return;                        // uniform per block
    const int half = lane >> 4;                 // 0: lanes 0-15, 1: lanes 16-31
    const int l15  = lane & 15;

    // ---- TDM: DMA the 16x128 fp32 A tile into LDS (2-DWORD pad per row) ----
    if (wave == 0) {
        const unsigned long long ga =
            (unsigned long long)(const void*)(A + (size_t)m0 * IN_DIM);
        const unsigned lds_base = (unsigned)(size_t)(void*)lds_A; // LDS byte offset
        const unsigned rows     = (unsigned)(M - m0);  // OOB rows read as zero

        u32x4 g0;
        g0.x = 1u;                                      // count=1, user D#
        g0.y = lds_base;                                // lds_addr[63:32]
        g0.z = (unsigned)(ga & 0xffffffffull);          // global_addr[31:0]
        g0.w = (unsigned)((ga >> 32) & 0x01ffffffull)   // global_addr[56:32]
             | (2u << 30);                              // type=2 ("image")

        i32x8 g1;
        g1[0] = (int)((2u << 16)        // data_size = 4 bytes
                    | (1u << 20)        // pad_enable
                    | (6u << 22)        // pad_interval: 128 DWORDs (one row)
                    | (1u << 25));      // pad_amount: 2 DWORDs
        g1[1] = (int)(((unsigned)IN_DIM & 0xffffu) << 16);        // tensor_dim0 lo16
        g1[2] = (int)((rows & 0xffffu) << 16);                    // dim0 hi16=0 | tensor_dim1 lo16
        g1[3] = (int)(((rows >> 16) & 0xffffu)                    // tensor_dim1 hi16
                    | ((unsigned)IN_DIM << 16));                  // tile_dim0 = 128
        g1[4] = 16;                                               // tile_dim1=16, tile_dim2=0
        g1[5] = IN_DIM;                                           // tensor_dim0_stride lo32
        g1[6] = 0;
        g1[7] = 0;

        i32x4 gz4 = {0, 0, 0, 0};             // groups 2/3 unused (2D tensor)
        i32x8 gz8 = {0, 0, 0, 0, 0, 0, 0, 0}; // extra group (clang-23 6-arg form)
        __builtin_amdgcn_tensor_load_to_lds(g0, g1, gz4, gz4, gz8, 0);
        __builtin_amdgcn_s_wait_tensorcnt(0);
    }
    __syncthreads();

    const int n0 = wave * 32;                   // two 16-wide col tiles
    const int aRow = l15 * LDS_PITCH;
    v8f acc0 = {};
    v8f acc1 = {};

    for (int k0 = 0; k0 < IN_DIM; k0 += 4) {
        const int ka = k0 + half * 2;
        // A fragment 16x4 from LDS (lanes0-15: K0/K1, lanes16-31: K2/K3)
        v2f a;
        a.x = lds_A[aRow + ka];
        a.y = lds_A[aRow + ka + 1];
        // B fragments 4x16 for the two column tiles
        v2f b0, b1;
        {
            const int col = n0 + l15;
            b0.x = Bsrc[(long)(ka + 0) * strideK + (long)col * strideO];
            b0.y = Bsrc[(long)(ka + 1) * strideK + (long)col * strideO];
        }
        {
            const int col = n0 + 16 + l15;
            b1.x = Bsrc[(long)(ka + 0) * strideK + (long)col * strideO];
            b1.y = Bsrc[(long)(ka + 1) * strideK + (long)col * strideO];
        }
        acc0 = __builtin_amdgcn_wmma_f32_16x16x4_f32(
                   false, a, false, b0, (short)0, acc0, false, false);
        acc1 = __builtin_amdgcn_wmma_f32_16x16x4_f32(
                   false, a, false, b1, (short)0, acc1, false, false);
    }

    // C/D layout: VGPR v -> row m0 + v + half*8, col n0 + l15
    if (m0 + 16 <= M) {                         // full tile: unguarded stores
        #pragma unroll
        for (int v = 0; v < 8; ++v) {
            float* o = out + (size_t)(m0 + v + half * 8) * OUT_DIM + n0 + l15;
            o[0]  = acc0[v];
            o[16] = acc1[v];
        }
    } else {                                    // ragged tail
        #pragma unroll
        for (int v = 0; v < 8; ++v) {
            const int row = m0 + v + half * 8;
            if (row < M) {
                float* o = out + (size_t)row * OUT_DIM + n0 + l15;
                o[0]  = acc0[v];
                o[16] = acc1[v];
            }
        }
    }
}

// -------------------------------------------------------------------------
// init: zero aggregation buffer (d_out), set max-keys to INT_MIN, denom to 0
// -------------------------------------------------------------------------
__global__ void gat_init(float* __restrict__ out, int* __restrict__ mkey,
                         float* __restrict__ denom, long n_out, long n_nh)
{
    const long stride = (long)gridDim.x * blockDim.x;
    for (long i = (long)blockIdx.x * blockDim.x + threadIdx.x; i < n_out; i += stride)
        out[i] = 0.0f;
    for (long i = (long)blockIdx.x * blockDim.x + threadIdx.x; i < n_nh; i += stride) {
        mkey[i]  = (int)0x80000000;
        denom[i] = 0.0f;
    }
}

// -------------------------------------------------------------------------
// per-(node,head) attention logits: a_src/a_dst = <h[n,h,:], att_*[h,:]>
// -------------------------------------------------------------------------
__global__ void gat_att(const float* __restrict__ hfeat,
                        const float* __restrict__ att_src,
                        const float* __restrict__ att_dst,
                        float* __restrict__ as_, float* __restrict__ ad_, int M)
{
    const int i = blockIdx.x * blockDim.x + threadIdx.x;
    if (i >= M * HEADS) return;
    const int n  = i / HEADS;
    const int hd = i % HEADS;
    const float* hp = hfeat + (long)n * OUT_DIM + hd * CH;
    const float* s1 = att_src + hd * CH;
    const float* s2 = att_dst + hd * CH;
    float ssum = 0.0f, dsum = 0.0f;
    #pragma unroll
    for (int c = 0; c < CH; ++c) {
        const float hv = hp[c];
        ssum += hv * s1[c];
        dsum += hv * s2[c];
    }
    as_[i] = ssum;
    ad_[i] = dsum;
}

__device__ __forceinline__ int float_orderable_key(float f) {
    int iv = __float_as_int(f);
    return (iv >= 0) ? iv : (iv ^ 0x7fffffff);   // monotonic float->int map
}
__device__ __forceinline__ float key_to_float(int k) {
    return __int_as_float((k >= 0) ? k : (k ^ 0x7fffffff));
}

// -------------------------------------------------------------------------
// edge pass A: segment max of LeakyReLU(a_src[src]+a_dst[dst]) per head
// self-loops implicit for e >= E (src = dst = e - E)
// -------------------------------------------------------------------------
__global__ void gat_edge_max(const int* __restrict__ ei,
                             const float* __restrict__ as_,
                             const float* __restrict__ ad_,
                             int* __restrict__ mkey, int E, int ET)
{
    const int e = blockIdx.x * blockDim.x + threadIdx.x;
    if (e >= ET) return;
    int s, d;
    if (e < E) { s = ei[e]; d = ei[E + e]; } else { s = d = e - E; }
    #pragma unroll
    for (int hd = 0; hd < HEADS; ++hd) {
        float v = as_[(long)s * HEADS + hd] + ad_[(long)d * HEADS + hd];
        v = (v > 0.0f) ? v : NEG_SLOPE * v;
        atomicMax(&mkey[(long)d * HEADS + hd], float_orderable_key(v));
    }
}

// -------------------------------------------------------------------------
// edge pass B: p = exp(e - max[dst]); store p, accumulate denom
// -------------------------------------------------------------------------
__global__ void gat_edge_expsum(const int* __restrict__ ei,
                                const float* __restrict__ as_,
                                const float* __restrict__ ad_,
                                const int* __restrict__ mkey,
                                float* __restrict__ pbuf,
                                float* __restrict__ denom, int E, int ET)
{
    const int e = blockIdx.x * blockDim.x + threadIdx.x;
    if (e >= ET) return;
    int s, d;
    if (e < E) { s = ei[e]; d = ei[E + e]; } else { s = d = e - E; }
    #pragma unroll
    for (int hd = 0; hd < HEADS; ++hd) {
        float v = as_[(long)s * HEADS + hd] + ad_[(long)d * HEADS + hd];
        v = (v > 0.0f) ? v : NEG_SLOPE * v;
        const float mx = key_to_float(mkey[(long)d * HEADS + hd]);
        const float p  = __expf(v - mx);
        pbuf[(long)e * HEADS + hd] = p;
        atomicAdd(&denom[(long)d * HEADS + hd], p);
    }
}

// -------------------------------------------------------------------------
// edge pass C: one wave per edge; scatter-add alpha * h[src] into out[dst]
// lanes 0-7 hold per-head coefficients, broadcast via wave32 shuffle
// -------------------------------------------------------------------------
__global__ __launch_bounds__(256) void gat_edge_aggregate(
    const int* __restrict__ ei, const float* __restrict__ hfeat,
    const float* __restrict__ pbuf, const float* __restrict__ denom,
    float* __restrict__ out, int E, int ET)
{
    const int wid  = blockIdx.x * (blockDim.x >> 5) + (threadIdx.x >> 5);
    const int lane = threadIdx.x & 31;
    if (wid >= ET) return;
    int s, d;
    if (wid < E) { s = ei[wid]; d = ei[E + wid]; } else { s = d = wid - E; }

    float cv = 0.0f;
    if (lane < HEADS) {
        const float pp = pbuf[(long)wid * HEADS + lane];
        const float dd = denom[(long)d * HEADS + lane];
        cv = pp / (dd + 1e-16f);
    }
    const float* hs = hfeat + (long)s * OUT_DIM;
    float*       od = out   + (long)d * OUT_DIM;
    #pragma unroll
    for (int hd = 0; hd < HEADS; ++hd) {
        const float coef = __shfl(cv, hd, 32);
        const float hv   = hs[hd * CH + lane];       // coalesced 128B gather
        atomicAdd(&od[hd * CH + lane], coef * hv);   // coalesced L2 atomics
    }
}

// -------------------------------------------------------------------------
// finalize: one wave per node — bias, LayerNorm (wave shuffle reduction),
// residual (identity + res_b), ELU; in-place on d_out
// -------------------------------------------------------------------------
__global__ __launch_bounds__(256) void gat_finalize(
    const float* __restrict__ ident, const float* __restrict__ bias,
    const float* __restrict__ gamma, const float* __restrict__ beta,
    const float* __restrict__ resb, float* __restrict__ out, int M)
{
    const int node = blockIdx.x * (blockDim.x >> 5) + (threadIdx.x >> 5);
    const int lane = threadIdx.x & 31;
    if (node >= M) return;

    float v[8];
    float s = 0.0f, s2 = 0.0f;
    #pragma unroll
    for (int j = 0; j < 8; ++j) {
        const int ch = j * 32 + lane;
        const float t = out[(long)node * OUT_DIM + ch] + bias[ch];
        v[j] = t; s += t; s2 += t * t;
    }
    #pragma unroll
    for (int off = 16; off >= 1; off >>= 1) {
        s  += __shfl_xor(s,  off, 32);
        s2 += __shfl_xor(s2, off, 32);
    }
    const float mean = s * (1.0f / OUT_DIM);
    const float var  = s2 * (1.0f / OUT_DIM) - mean * mean;
    const float rinv = rsqrtf(var + LN_EPS);
    #pragma unroll
    for (int j = 0; j < 8; ++j) {
        const int ch = j * 32 + lane;
        const float y  = gamma[ch] * (v[j] - mean) * rinv + beta[ch];
        const float id = ident[(long)node * OUT_DIM + ch] + resb[ch];
        const float z  = y + id;
        out[(long)node * OUT_DIM + ch] = (z > 0.0f) ? z : (__expf(z) - 1.0f);
    }
}

// -------------------------------------------------------------------------
extern "C" void kernel_launch(void* const* d_in, const int* in_sizes, int n_in,
                              void* d_out, int out_size, void* d_ws, size_t ws_size,
                              hipStream_t stream)
{
    const float* x       = (const float*)d_in[0];
    const int*   ei      = (const int*)  d_in[1];
    const float* W       = (const float*)d_in[2];
    const float* att_src = (const float*)d_in[3];
    const float* att_dst = (const float*)d_in[4];
    const float* bias    = (const float*)d_in[5];
    const float* gamma   = (const float*)d_in[6];
    const float* beta    = (const float*)d_in[7];
    const float* resW    = (const float*)d_in[8];
    const float* resb    = (const float*)d_in[9];
    float* out = (float*)d_out;

    const int N  = in_sizes[0] / IN_DIM;
    const int E  = in_sizes[1] / 2;
    const int ET = E + N;

    // workspace layout (floats)
    float* ws    = (float*)d_ws;
    size_t off   = 0;
    float* hfeat = ws + off; off += (size_t)N * OUT_DIM;
    float* ident = ws + off; off += (size_t)N * OUT_DIM;
    float* as_   = ws + off; off += (size_t)N * HEADS;
    float* ad_   = ws + off; off += (size_t)N * HEADS;
    int*   mkey  = (int*)(ws + off); off += (size_t)N * HEADS;
    float* denom = ws + off; off += (size_t)N * HEADS;
    float* pbuf  = ws + off; off += (size_t)ET * HEADS;

    const int mtiles = (N + 15) / 16;

    gat_init<<<(int)(((long)N * OUT_DIM + 255) / 256), 256, 0, stream>>>(
        out, mkey, denom, (long)N * OUT_DIM, (long)N * HEADS);

    // h = x @ W           (B(k,o) = W[k*256 + o])
    gat_gemm_f32_wmma<<<mtiles, 256, 0, stream>>>(x, W, hfeat, N, OUT_DIM, 1);
    // identity = x @ W_r^T (B(k,o) = res_W[o*128 + k])
    gat_gemm_f32_wmma<<<mtiles, 256, 0, stream>>>(x, resW, ident, N, 1, IN_DIM);

    gat_att<<<(N * HEADS + 255) / 256, 256, 0, stream>>>(
        hfeat, att_src, att_dst, as_, ad_, N);

    gat_edge_max<<<(ET + 255) / 256, 256, 0, stream>>>(ei, as_, ad_, mkey, E, ET);
    gat_edge_expsum<<<(ET + 255) / 256, 256, 0, stream>>>(
        ei, as_, ad_, mkey, pbuf, denom, E, ET);
    gat_edge_aggregate<<<(ET + 7) / 8, 256, 0, stream>>>(
        ei, hfeat, pbuf, denom, out, E, ET);

    gat_finalize<<<(N + 7) / 8, 256, 0, stream>>>(
        ident, bias, gamma, beta, resb, out, N);
}